// RSEncoderLayer_23416161697928
// MI455X (gfx1250) — compile-verified
//
#include <hip/hip_runtime.h>

#define N_NODES   50000
#define N_EDGES   1600000
#define IN_DIM    256
#define OUT_DIM   128
#define T_STEPS   8

typedef float v2f __attribute__((ext_vector_type(2)));
typedef float v8f __attribute__((ext_vector_type(8)));

// ---------------------------------------------------------------------------
// Kernel 1: h = x @ W  via v_wmma_f32_16x16x4_f32 (wave32).
// Block = 256 threads = 8 waves. Each block computes a 16-row slab of h;
// wave w computes the 16-wide column tile [16w, 16w+16).
// x tile (16 x 256 f32) staged in LDS with padded stride to kill bank
// conflicts (stride 258 -> bank step 2 per row, conflict-free for 16 rows).
// ---------------------------------------------------------------------------
#define A_STRIDE 258

__global__ __launch_bounds__(256) void gemm_wmma_f32(
    const float* __restrict__ x, const float* __restrict__ W,
    float* __restrict__ h) {
  __shared__ float As[16 * A_STRIDE];

  const int rowBase = blockIdx.x * 16;  // 50000 = 3125 * 16, exact

  // Cooperative float4 load of x[rowBase:rowBase+16, 0:256] into LDS.
  for (int i = threadIdx.x; i < 16 * (IN_DIM / 4); i += 256) {
    const int r  = i / (IN_DIM / 4);
    const int c4 = i % (IN_DIM / 4);
    const float4 v =
        ((const float4*)(x + (size_t)(rowBase + r) * IN_DIM))[c4];
    float* dst = &As[r * A_STRIDE + c4 * 4];
    dst[0] = v.x; dst[1] = v.y; dst[2] = v.z; dst[3] = v.w;
  }
  __syncthreads();

  const int wave    = threadIdx.x >> 5;
  const int lane    = threadIdx.x & 31;
  const int half    = lane >> 4;   // 0: K pair {0,1}, 1: K pair {2,3}
  const int l16     = lane & 15;   // A: row M, B: col N
  const int colBase = wave * 16;

  v8f c = {};
  const float* arow = &As[l16 * A_STRIDE];

  #pragma unroll 4
  for (int k = 0; k < IN_DIM; k += 4) {
    const int ka = k + half * 2;
    v2f a;
    a.x = arow[ka];
    a.y = arow[ka + 1];
    v2f b;
    b.x = W[(size_t)ka * OUT_DIM + colBase + l16];
    b.y = W[(size_t)(ka + 1) * OUT_DIM + colBase + l16];
    // 8 args: (neg_a, A, neg_b, B, c_mod, C, reuse_a, reuse_b)
    c = __builtin_amdgcn_wmma_f32_16x16x4_f32(false, a, false, b,
                                              (short)0, c, false, false);
  }

  // D layout: VGPR v holds row M = v + 8*half, col N = l16.
  #pragma unroll
  for (int v = 0; v < 8; ++v) {
    const int m = v + 8 * half;
    h[(size_t)(rowBase + m) * OUT_DIM + colBase + l16] = c[v];
  }
}

// ---------------------------------------------------------------------------
// Degree / normalization kernels.
// ---------------------------------------------------------------------------
__global__ void init_deg(float* __restrict__ deg) {
  const int i = blockIdx.x * blockDim.x + threadIdx.x;
  if (i < N_NODES) deg[i] = 1.0f;  // self-loop contributes 1 to every node
}

__global__ void accum_deg(const int* __restrict__ ei, float* __restrict__ deg) {
  const int e = blockIdx.x * blockDim.x + threadIdx.x;
  if (e < N_EDGES) atomicAdd(&deg[ei[N_EDGES + e]], 1.0f);  // dst row
}

__global__ void deg_to_dinv(float* __restrict__ deg) {
  const int i = blockIdx.x * blockDim.x + threadIdx.x;
  if (i < N_NODES) deg[i] = rsqrtf(deg[i]);  // deg >= 1 always (self-loop)
}

// ---------------------------------------------------------------------------
// agg init with self-loop term: agg[i,:] = h[i,:] * dinv[i]^2
// ---------------------------------------------------------------------------
__global__ void self_loop_init(const float* __restrict__ h,
                               const float* __restrict__ dinv,
                               float* __restrict__ agg) {
  const int i = blockIdx.x * blockDim.x + threadIdx.x;  // float4 index
  const int total4 = N_NODES * (OUT_DIM / 4);
  if (i >= total4) return;
  const int node = i / (OUT_DIM / 4);
  const float d = dinv[node];
  const float n = d * d;
  float4 hv = ((const float4*)h)[i];
  hv.x *= n; hv.y *= n; hv.z *= n; hv.w *= n;
  ((float4*)agg)[i] = hv;
}

// ---------------------------------------------------------------------------
// Edge scatter: one wave per edge, 4 consecutive floats per lane.
// h (25.6 MB) and agg (25.6 MB) are L2-resident (192 MB L2), so both the
// random gather of h[src,:] and the float atomics to agg[dst,:] stay on-chip.
// ---------------------------------------------------------------------------
__global__ __launch_bounds__(256) void edge_scatter(
    const int* __restrict__ ei, const float* __restrict__ h,
    const float* __restrict__ dinv, float* __restrict__ agg) {
  const int lane = threadIdx.x & 31;
  const long long edge =
      ((long long)blockIdx.x * blockDim.x + threadIdx.x) >> 5;
  if (edge >= N_EDGES) return;
  const int src = ei[edge];
  const int dst = ei[N_EDGES + edge];
  const float norm = dinv[src] * dinv[dst];
  const float4 hv = ((const float4*)(h + (size_t)src * OUT_DIM))[lane];
  float* out = agg + (size_t)dst * OUT_DIM + lane * 4;
  atomicAdd(out + 0, hv.x * norm);
  atomicAdd(out + 1, hv.y * norm);
  atomicAdd(out + 2, hv.z * norm);
  atomicAdd(out + 3, hv.w * norm);
}

// ---------------------------------------------------------------------------
// Fused LIF neuron loop, T=8, fully unrolled. Hm = (z+u)/2 (TAU=2),
// o = (Hm >= 1), z = Hm - o. Writes o_seq then z_seq (concatenated in d_out).
// HBM-store bound: 410 MB of float4 streaming stores.
// ---------------------------------------------------------------------------
__global__ __launch_bounds__(256) void snn_scan(const float* __restrict__ agg,
                                                float* __restrict__ out) {
  const int i = blockIdx.x * blockDim.x + threadIdx.x;  // float4 index
  const int total4 = N_NODES * (OUT_DIM / 4);
  if (i >= total4) return;

  float4 u = ((const float4*)agg)[i];
  u.x *= 0.1f; u.y *= 0.1f; u.z *= 0.1f; u.w *= 0.1f;  // STEP_SIZE
  float4 z = make_float4(0.f, 0.f, 0.f, 0.f);

  const size_t stride4 = (size_t)N_NODES * (OUT_DIM / 4);  // per-step, float4
  float4* o_out = (float4*)out;
  float4* z_out = o_out + (size_t)T_STEPS * stride4;

  #pragma unroll
  for (int t = 0; t < T_STEPS; ++t) {
    float4 Hm;
    Hm.x = (z.x + u.x) * 0.5f;
    Hm.y = (z.y + u.y) * 0.5f;
    Hm.z = (z.z + u.z) * 0.5f;
    Hm.w = (z.w + u.w) * 0.5f;
    float4 o;
    o.x = (Hm.x >= 1.0f) ? 1.0f : 0.0f;
    o.y = (Hm.y >= 1.0f) ? 1.0f : 0.0f;
    o.z = (Hm.z >= 1.0f) ? 1.0f : 0.0f;
    o.w = (Hm.w >= 1.0f) ? 1.0f : 0.0f;
    z.x = Hm.x - o.x;
    z.y = Hm.y - o.y;
    z.z = Hm.z - o.z;
    z.w = Hm.w - o.w;
    o_out[t * stride4 + i] = o;
    z_out[t * stride4 + i] = z;
  }
}

// ---------------------------------------------------------------------------
extern "C" void kernel_launch(void* const* d_in, const int* in_sizes, int n_in,
                              void* d_out, int out_size, void* d_ws,
                              size_t ws_size, hipStream_t stream) {
  const float* x  = (const float*)d_in[0];   // [50000, 256]
  const float* W  = (const float*)d_in[1];   // [256, 128]
  const int*   ei = (const int*)d_in[2];     // [2, 1600000]
  float* out = (float*)d_out;                // o_seq | z_seq, each [8,50000,128]

  // Workspace layout (~51.4 MB):
  float* h   = (float*)d_ws;                         // [50000, 128]
  float* agg = h + (size_t)N_NODES * OUT_DIM;        // [50000, 128]
  float* deg = agg + (size_t)N_NODES * OUT_DIM;      // [50000] (deg -> dinv)

  const int total4 = N_NODES * (OUT_DIM / 4);

  gemm_wmma_f32<<<N_NODES / 16, 256, 0, stream>>>(x, W, h);

  init_deg<<<(N_NODES + 255) / 256, 256, 0, stream>>>(deg);
  accum_deg<<<(N_EDGES + 255) / 256, 256, 0, stream>>>(ei, deg);
  deg_to_dinv<<<(N_NODES + 255) / 256, 256, 0, stream>>>(deg);

  self_loop_init<<<(total4 + 255) / 256, 256, 0, stream>>>(h, deg, agg);

  const long long scatter_threads = (long long)N_EDGES * 32;
  edge_scatter<<<(unsigned)((scatter_threads + 255) / 256), 256, 0, stream>>>(
      ei, h, deg, agg);

  snn_scan<<<(total4 + 255) / 256, 256, 0, stream>>>(agg, out);
}